// Net_15513421873302
// MI455X (gfx1250) — compile-verified
//
#include <hip/hip_runtime.h>

typedef __attribute__((ext_vector_type(2))) float v2f;
typedef __attribute__((ext_vector_type(8))) float v8f;

#define NN 50000
#define NE 800000
#define F_IN 100
#define F_HID 128
#define F_OUT 47
#define LN_EPS 1e-5f

// ---------------- degree ----------------
__global__ void deg_init_kernel(float* __restrict__ deg, int n) {
    int i = blockIdx.x * blockDim.x + threadIdx.x;
    if (i < n) deg[i] = 1.0f;  // self loop
}

__global__ void deg_count_kernel(const int* __restrict__ row, float* __restrict__ deg, int e) {
    int i = blockIdx.x * blockDim.x + threadIdx.x;
    if (i < e) atomicAdd(&deg[row[i]], 1.0f);
}

// ---------------- aggregation ----------------
// agg[i,:] = (2/deg[i]) * h[i,:]   (self loop with diag_lambda=1 enhancement)
__global__ void agg_init_kernel(const float* __restrict__ h, const float* __restrict__ deg,
                                float* __restrict__ agg, int f) {
    int node = blockIdx.x;
    float s = 2.0f / deg[node];
    const float* src = h + (long)node * f;
    float* dst = agg + (long)node * f;
    for (int ft = threadIdx.x; ft < f; ft += blockDim.x)
        dst[ft] = s * src[ft];
}

// one wave per edge: agg[row] += rsqrt(deg[row])*rsqrt(deg[col]) * h[col]
// lanes each own a float4 chunk (f is a multiple of 4; rows are 16B aligned).
__global__ void scatter_kernel(const int* __restrict__ row, const int* __restrict__ col,
                               const float* __restrict__ deg, const float* __restrict__ h,
                               float* __restrict__ agg, int e, int f) {
    int wave = (blockIdx.x * blockDim.x + threadIdx.x) >> 5;
    int lane = threadIdx.x & 31;
    if (wave >= e) return;
    int r = row[wave], c = col[wave];
    float w = rsqrtf(deg[r]) * rsqrtf(deg[c]);
    const float4* src = (const float4*)(h + (long)c * f);
    float* dst = agg + (long)r * f;
    int nchunks = f >> 2;                       // 25 (F=100) or 32 (F=128)
    for (int q = lane; q < nchunks; q += 32) {
        float4 v = src[q];
        int ft = q << 2;
        atomicAdd(&dst[ft + 0], w * v.x);
        atomicAdd(&dst[ft + 1], w * v.y);
        atomicAdd(&dst[ft + 2], w * v.z);
        atomicAdd(&dst[ft + 3], w * v.w);
    }
}

// ---------------- W transpose + zero-pad: Wt[foPad, 2f] = W[2f, fo]^T ----------------
__global__ void transpose_pad_kernel(const float* __restrict__ W, float* __restrict__ Wt,
                                     int twoF, int fo) {
    int j = blockIdx.x;                         // 0 .. foPad-1
    float* dst = Wt + (long)j * twoF;
    if (j < fo) {
        for (int k = threadIdx.x; k < twoF; k += blockDim.x)
            dst[k] = W[(long)k * fo + j];
    } else {
        for (int k = threadIdx.x; k < twoF; k += blockDim.x)
            dst[k] = 0.0f;                      // pad columns contribute zeros
    }
}

// ---------------- GEMM: out = [A1 | A2] @ W  via V_WMMA_F32_16X16X4_F32 ----------------
// A1,A2: [n,f] row-major; Wt: [foPad, 2f] (transposed, zero-padded); out: [n,fo].
// One wave computes one 16x16 tile; 8 waves (8 row-tiles) per 256-thread block.
// Inner loop: one b64 load for A, one b64 load for B, one WMMA.
__global__ void gemm_wmma_kernel(const float* __restrict__ A1,
                                 const float* __restrict__ A2,
                                 const float* __restrict__ Wt,
                                 float* __restrict__ out,
                                 int n, int f, int fo) {
    int lane = threadIdx.x & 31;
    int wave = threadIdx.x >> 5;
    int rowBase = (blockIdx.x * 8 + wave) * 16;
    int colBase = blockIdx.y * 16;
    if (rowBase >= n) return;                   // wave-uniform: EXEC all-ones at WMMA

    int m     = lane & 15;                      // A row within tile / B,D column
    int khalf = (lane >> 4) << 1;               // lanes 0-15: K {0,1}; lanes 16-31: K {2,3}
    int aRow  = rowBase + m;
    if (aRow >= n) aRow = n - 1;                // clamp loads; stores are bounds-checked
    int bCol  = colBase + m;                    // < foPad by construction (pad rows = 0)

    const float* aBase1 = A1 + (long)aRow * f + khalf;
    const float* aBase2 = A2 + (long)aRow * f + khalf;
    const float* bBase  = Wt + (long)bCol * (2 * f) + khalf;

    v8f acc = {};
    // panel 1: agg half (W rows [0,f))
    #pragma unroll 4
    for (int k = 0; k < f; k += 4) {
        v2f a = *(const v2f*)(aBase1 + k);
        v2f b = *(const v2f*)(bBase + k);
        acc = __builtin_amdgcn_wmma_f32_16x16x4_f32(false, a, false, b,
                                                    (short)0, acc, false, false);
    }
    // panel 2: x/h half (W rows [f,2f))
    #pragma unroll 4
    for (int k = 0; k < f; k += 4) {
        v2f a = *(const v2f*)(aBase2 + k);
        v2f b = *(const v2f*)(bBase + f + k);
        acc = __builtin_amdgcn_wmma_f32_16x16x4_f32(false, a, false, b,
                                                    (short)0, acc, false, false);
    }
    // D layout: VGPR r -> row (r + 8*(lane>=16)), col = lane&15
    if (bCol < fo) {
        int rOff = (lane >> 4) * 8;
        #pragma unroll
        for (int r = 0; r < 8; ++r) {
            int orow = rowBase + rOff + r;
            if (orow < n) out[(long)orow * fo + bCol] = acc[r];
        }
    }
}

// ---------------- layer norm (+ optional relu), one wave per row ----------------
__global__ void layernorm_kernel(const float* __restrict__ in, float* __restrict__ outp,
                                 int n, int f, int doRelu) {
    int wave = (blockIdx.x * blockDim.x + threadIdx.x) >> 5;
    int lane = threadIdx.x & 31;
    if (wave >= n) return;
    const float* src = in + (long)wave * f;
    float v[4];                                 // f <= 128
    float s = 0.f, ss = 0.f;
    int cnt = 0;
    for (int ft = lane; ft < f; ft += 32) {
        float val = src[ft];
        v[cnt++] = val;
        s += val; ss += val * val;
    }
    for (int msk = 16; msk >= 1; msk >>= 1) {
        s  += __shfl_xor(s,  msk, 32);
        ss += __shfl_xor(ss, msk, 32);
    }
    float mu  = s / (float)f;
    float var = ss / (float)f - mu * mu;
    float inv = rsqrtf(var + LN_EPS);
    float* dst = outp + (long)wave * f;
    cnt = 0;
    for (int ft = lane; ft < f; ft += 32) {
        float y = (v[cnt++] - mu) * inv;
        if (doRelu) y = fmaxf(y, 0.f);
        dst[ft] = y;
    }
}

extern "C" void kernel_launch(void* const* d_in, const int* in_sizes, int n_in,
                              void* d_out, int out_size, void* d_ws, size_t ws_size,
                              hipStream_t stream) {
    const float* x   = (const float*)d_in[0];           // [N, 100]
    const int*   ei  = (const int*)d_in[1];             // [2, E]
    const float* W1  = (const float*)d_in[2];           // [200,128]
    const float* W2  = (const float*)d_in[3];           // [256,128]
    const float* W3  = (const float*)d_in[4];           // [256, 47]
    float*       yo  = (float*)d_out;                   // [N, 47]
    const int* row = ei;
    const int* col = ei + NE;

    float* ws   = (float*)d_ws;
    float* deg  = ws;                                   // N
    float* agg  = ws + ((NN + 255) & ~255);             // N*128
    float* gout = agg + (long)NN * 128;                 // N*128
    float* hA   = gout + (long)NN * 128;                // N*128
    float* hB   = hA + (long)NN * 128;                  // N*128
    float* Wt   = hB + (long)NN * 128;                  // 128*256 max (transposed W)

    const int TB = 256;
    dim3 blk(TB);
    int edgeWaves = (NE * 32 + TB - 1) / TB;
    int rowWaves  = (NN * 32 + TB - 1) / TB;
    int rowTiles8 = (NN + 127) / 128;                   // 391 blocks of 8 row-tiles

    // degrees (with self loops)
    deg_init_kernel<<<(NN + TB - 1) / TB, blk, 0, stream>>>(deg, NN);
    deg_count_kernel<<<(NE + TB - 1) / TB, blk, 0, stream>>>(row, deg, NE);

    // ---- layer 1: x (F=100) -> hA (128), relu ----
    agg_init_kernel<<<NN, 128, 0, stream>>>(x, deg, agg, F_IN);
    scatter_kernel<<<edgeWaves, blk, 0, stream>>>(row, col, deg, x, agg, NE, F_IN);
    transpose_pad_kernel<<<128, blk, 0, stream>>>(W1, Wt, 2 * F_IN, F_HID);
    gemm_wmma_kernel<<<dim3(rowTiles8, F_HID / 16), blk, 0, stream>>>(agg, x, Wt, gout, NN, F_IN, F_HID);
    layernorm_kernel<<<rowWaves, blk, 0, stream>>>(gout, hA, NN, F_HID, 1);

    // ---- layer 2: hA (128) -> hB (128), relu ----
    agg_init_kernel<<<NN, 128, 0, stream>>>(hA, deg, agg, F_HID);
    scatter_kernel<<<edgeWaves, blk, 0, stream>>>(row, col, deg, hA, agg, NE, F_HID);
    transpose_pad_kernel<<<128, blk, 0, stream>>>(W2, Wt, 2 * F_HID, F_HID);
    gemm_wmma_kernel<<<dim3(rowTiles8, F_HID / 16), blk, 0, stream>>>(agg, hA, Wt, gout, NN, F_HID, F_HID);
    layernorm_kernel<<<rowWaves, blk, 0, stream>>>(gout, hB, NN, F_HID, 1);

    // ---- layer 3: hB (128) -> out (47), no relu ----
    agg_init_kernel<<<NN, 128, 0, stream>>>(hB, deg, agg, F_HID);
    scatter_kernel<<<edgeWaves, blk, 0, stream>>>(row, col, deg, hB, agg, NE, F_HID);
    transpose_pad_kernel<<<48, blk, 0, stream>>>(W3, Wt, 2 * F_HID, F_OUT);
    gemm_wmma_kernel<<<dim3(rowTiles8, 3), blk, 0, stream>>>(agg, hB, Wt, gout, NN, F_HID, F_OUT);
    layernorm_kernel<<<rowWaves, blk, 0, stream>>>(gout, yo, NN, F_HID == 0 ? 0 : F_OUT, 0);
}